// CoordinateBasedDownsampling_70007966925288
// MI455X (gfx1250) — compile-verified
//
#include <hip/hip_runtime.h>
#include <math.h>

typedef __attribute__((ext_vector_type(2))) float v2f;
typedef __attribute__((ext_vector_type(8))) float v8f;

#define DD 256

// ---------------------------------------------------------------- init
__global__ void k_init(unsigned* cmm, unsigned* counts, unsigned* minidx,
                       unsigned* rankbuf, unsigned* nu, float* rankrecip,
                       float* outbuf, int B, int nc3, long n_rank_f, long n_out) {
  long i = (long)blockIdx.x * blockDim.x + threadIdx.x;
  long stride = (long)gridDim.x * blockDim.x;
  for (long j = i; j < n_out; j += stride) outbuf[j] = 0.0f;
  long ncnt = (long)B * nc3;
  for (long j = i; j < ncnt; j += stride) {
    counts[j] = 0u; minidx[j] = 0xFFFFFFFFu; rankbuf[j] = 0u;
  }
  for (long j = i; j < n_rank_f; j += stride) rankrecip[j] = 0.0f;
  if (i < (long)B * 6) cmm[i] = ((i % 6) < 3) ? 0x7F7FFFFFu : 0u; // +FLT_MAX / 0
  if (i < B) nu[i] = 0u;
}

// ---------------------------------------------------------------- coord min/max
__global__ void k_minmax(const float* __restrict__ coords, unsigned* cmm, int N) {
  __shared__ unsigned smin[3][256];
  __shared__ unsigned smax[3][256];
  int t = threadIdx.x;
  long token = (long)blockIdx.x * 256 + t;
  int b = (int)(token / N);
  for (int j = 0; j < 3; ++j) {
    unsigned u = __float_as_uint(coords[token * 3 + j]); // coords >= 0
    smin[j][t] = u; smax[j][t] = u;
  }
  __syncthreads();
  for (int s = 128; s > 0; s >>= 1) {
    if (t < s)
      for (int j = 0; j < 3; ++j) {
        smin[j][t] = min(smin[j][t], smin[j][t + s]);
        smax[j][t] = max(smax[j][t], smax[j][t + s]);
      }
    __syncthreads();
  }
  if (t < 3)      atomicMin(&cmm[b * 6 + t], smin[t][0]);
  else if (t < 6) atomicMax(&cmm[b * 6 + t], smax[t - 3][0]);
}

// ---------------------------------------------------------------- cell ids + counts + first index
__global__ void k_cellid(const float* __restrict__ coords, const unsigned* __restrict__ cmm,
                         unsigned* counts, unsigned* minidx, unsigned* flatid,
                         int N, int cells, int nc3) {
  long token = (long)blockIdx.x * 256 + threadIdx.x;
  int b = (int)(token / N);
  int n = (int)(token % N);
  int g[3];
  for (int j = 0; j < 3; ++j) {
    float cmin = __uint_as_float(cmm[b * 6 + j]);
    float cmax = __uint_as_float(cmm[b * 6 + 3 + j]);
    float crange = cmax - cmin + 1e-6f;
    float nrm = (coords[token * 3 + j] - cmin) / crange;
    int gi = (int)(nrm * (float)cells);              // trunc toward zero (nonneg)
    g[j] = min(max(gi, 0), cells - 1);
  }
  unsigned flat = (unsigned)(g[0] * cells * cells + g[1] * cells + g[2]);
  flatid[token] = flat;
  atomicAdd(&counts[(long)b * nc3 + flat], 1u);
  atomicMin(&minidx[(long)b * nc3 + flat], (unsigned)n);
}

// ---------------------------------------------------------------- per-batch compact scan (np.unique order)
__global__ void k_scan(const unsigned* __restrict__ counts, unsigned* rankbuf,
                       unsigned* nu, int nc3) {
  __shared__ unsigned sdata[256];
  int t = threadIdx.x;
  int b = blockIdx.x;
  const unsigned* cnt = counts + (long)b * nc3;
  unsigned* rnk = rankbuf + (long)b * nc3;
  unsigned base = 0;
  for (int c0 = 0; c0 < nc3; c0 += 256) {
    int idx = c0 + t;
    unsigned occ = (idx < nc3 && cnt[idx] != 0u) ? 1u : 0u;
    sdata[t] = occ;
    __syncthreads();
    for (int s = 1; s < 256; s <<= 1) {
      unsigned v = (t >= s) ? sdata[t - s] : 0u;
      __syncthreads();
      sdata[t] += v;
      __syncthreads();
    }
    unsigned incl = sdata[t];
    if (idx < nc3) rnk[idx] = base + incl - occ;   // exclusive prefix
    unsigned tot = sdata[255];
    __syncthreads();
    base += tot;
  }
  if (t == 0) nu[b] = base;
}

// ---------------------------------------------------------------- emit mask / coords / 1/count
__global__ void k_emit(const unsigned* __restrict__ counts, const unsigned* __restrict__ minidx,
                       const unsigned* __restrict__ rankbuf, const float* __restrict__ coords,
                       float* pcoords, float* masks, float* rankrecip,
                       int N, int nc3, int max_len) {
  long i = (long)blockIdx.x * 256 + threadIdx.x;     // over B*nc3
  if (i >= (long)gridDim.x * 256) return;
  int b = (int)(i / nc3);
  if (i >= (long)(b + 1) * nc3) return;              // safety on tail
  unsigned cnt = counts[i];
  if (cnt == 0u) return;
  unsigned r = rankbuf[i];
  if ((int)r >= max_len) return;
  unsigned mi = minidx[i];
  long row = (long)b * max_len + r;
  masks[row] = 1.0f;
  rankrecip[row] = 1.0f / (float)cnt;
  for (int j = 0; j < 3; ++j)
    pcoords[row * 3 + j] = coords[((long)b * N + mi) * 3 + j];
}

// ---------------------------------------------------------------- scatter-add x rows into out region
__global__ void k_scatter(const float* __restrict__ x, const unsigned* __restrict__ flatid,
                          const unsigned* __restrict__ rankbuf, float* outbuf,
                          int N, int nc3, int max_len) {
  long token = blockIdx.x;                           // B*N blocks, 256 threads
  int t = threadIdx.x;
  int b = (int)(token / N);
  unsigned flat = flatid[token];
  unsigned r = rankbuf[(long)b * nc3 + flat];
  atomicAdd(&outbuf[((long)b * max_len + r) * DD + t], x[token * DD + t]);
}

// ---------------------------------------------------------------- mean + GEMM (WMMA f32 16x16x4) + mask, in place
__global__ __launch_bounds__(512)
void k_gemm(const float* __restrict__ W, const float* __restrict__ bproj,
            const float* __restrict__ rankrecip, const float* __restrict__ masks,
            float* outbuf, int max_len, int tiles_m) {
  __shared__ float lA[16 * 257];                     // stride 257 -> conflict-free column reads
  int tid = threadIdx.x;
  int tileM = blockIdx.x % tiles_m;
  int b = blockIdx.x / tiles_m;
  int rowbase = tileM * 16;

  // Stage mean-scaled 16x256 A tile into LDS (sums * 1/count; 0 beyond valid rows).
  for (int i = tid; i < 16 * DD; i += 512) {
    int row = i >> 8, col = i & 255;
    int m = rowbase + row;
    float v = 0.0f;
    if (m < max_len) {
      long rix = (long)b * max_len + m;
      v = outbuf[rix * DD + col] * rankrecip[rix];
    }
    lA[row * 257 + col] = v;
  }
  __syncthreads();                                   // all global A reads complete here

  int wave = tid >> 5;                               // 16 waves, one 16-wide N tile each
  int lane = tid & 31;
  int half = lane >> 4;                              // 0: K 0..1 / 1: K 2..3 (ISA 16x4 A layout)
  int l15 = lane & 15;
  int nbase = wave * 16;
  int n = nbase + l15;

  v8f c;
  float bias = bproj[n];                             // C layout: N = lane&15 for every VGPR
  for (int v = 0; v < 8; ++v) c[v] = bias;

  const float* wrow = W + (long)n * DD;              // out[l,n] = sum_e A[l,e] * W[n,e]
  __builtin_prefetch(wrow, 0, 1);
  int arow = l15;
  int koff = half * 2;

  for (int k = 0; k < DD; k += 4) {
    v2f a, bm;
    a.x = lA[arow * 257 + k + koff];
    a.y = lA[arow * 257 + k + koff + 1];
    bm.x = wrow[k + koff];
    bm.y = wrow[k + koff + 1];
    // D = A(16x4,f32) * B(4x16,f32) + C ; emits v_wmma_f32_16x16x4_f32
    c = __builtin_amdgcn_wmma_f32_16x16x4_f32(false, a, false, bm,
                                              (short)0, c, false, false);
  }

  // D layout: component v -> M = rowbase + v + half*8 ; N = lane&15. Apply mask, store in place.
  for (int v = 0; v < 8; ++v) {
    int m = rowbase + v + half * 8;
    if (m < max_len) {
      long rix = (long)b * max_len + m;
      outbuf[rix * DD + n] = c[v] * masks[rix];
    }
  }
}

// ---------------------------------------------------------------- host
extern "C" void kernel_launch(void* const* d_in, const int* in_sizes, int n_in,
                              void* d_out, int out_size, void* d_ws, size_t ws_size,
                              hipStream_t stream) {
  const float* x      = (const float*)d_in[0];   // (B,N,256)
  const float* coords = (const float*)d_in[1];   // (B,N,3)
  const float* W      = (const float*)d_in[2];   // (256,256)
  const float* bproj  = (const float*)d_in[3];   // (256)

  const int B = 8;
  const int D = DD;
  const long BN = (long)in_sizes[0] / D;
  const int N = (int)(BN / B);                   // 16384
  int cells = (int)cbrt((double)N / 2.0);        // FACTOR=2 -> 20
  if (cells < 2) cells = 2;
  const int nc3 = cells * cells * cells;         // 8000
  const int max_len = out_size / (B * (D + 3 + 1));

  float* outbuf  = (float*)d_out;                         // B*max_len*256 (also accumulator)
  float* pcoords = outbuf + (long)B * max_len * D;        // B*max_len*3
  float* masks   = pcoords + (long)B * max_len * 3;       // B*max_len

  unsigned* ws       = (unsigned*)d_ws;
  unsigned* cmm      = ws;                                // B*6   (min|max uint-bits)
  unsigned* counts   = cmm + B * 6;                       // B*nc3
  unsigned* minidx   = counts + (long)B * nc3;            // B*nc3
  unsigned* rankbuf  = minidx + (long)B * nc3;            // B*nc3
  unsigned* nu       = rankbuf + (long)B * nc3;           // B
  unsigned* flatid   = nu + B;                            // B*N
  float*    rankrecip= (float*)(flatid + (long)B * N);    // B*max_len

  const long n_out = (long)out_size;
  const long n_rank_f = (long)B * max_len;

  k_init<<<4096, 256, 0, stream>>>(cmm, counts, minidx, rankbuf, nu, rankrecip,
                                   outbuf, B, nc3, n_rank_f, n_out);

  const int tokBlocks = (int)(((long)B * N) / 256);       // 512
  k_minmax<<<tokBlocks, 256, 0, stream>>>(coords, cmm, N);

  k_cellid<<<tokBlocks, 256, 0, stream>>>(coords, cmm, counts, minidx, flatid,
                                          N, cells, nc3);

  k_scan<<<B, 256, 0, stream>>>(counts, rankbuf, nu, nc3);

  const int emitBlocks = (int)(((long)B * nc3 + 255) / 256);
  k_emit<<<emitBlocks, 256, 0, stream>>>(counts, minidx, rankbuf, coords,
                                         pcoords, masks, rankrecip, N, nc3, max_len);

  k_scatter<<<(int)((long)B * N), 256, 0, stream>>>(x, flatid, rankbuf, outbuf,
                                                    N, nc3, max_len);

  const int tiles_m = (max_len + 15) / 16;
  k_gemm<<<B * tiles_m, 512, 0, stream>>>(W, bproj, rankrecip, masks,
                                          outbuf, max_len, tiles_m);
}